// TemperatureRouter_32908039421952
// MI455X (gfx1250) — compile-verified
//
#include <hip/hip_runtime.h>
#include <math.h>

typedef __attribute__((ext_vector_type(2))) float v2f;
typedef __attribute__((ext_vector_type(8))) float v8f;

#define D_MODEL 2048
#define N_EXP   64
#define N_TOK   16384
#define TOK_PER_BLOCK 64

// flat output offsets (reference tuple order)
#define OFF_W    0            // top_k_weights  [N_TOK,2]
#define OFF_I    32768        // top_k_indices  [N_TOK,2] (as float)
#define OFF_P    65536        // router_probs   [N_TOK,64]
#define OFF_ENT  1114112      // routing_entropy scalar
#define OFF_CONF 1114113      // selection_confidence scalar
#define OFF_UTIL 1114114      // expert_utilization [64]

// ws layout: ws[0]=entropy sum, ws[1]=confidence sum, ws[2..65]=expert counts
__global__ void router_init_kernel(float* __restrict__ ws) {
    if (threadIdx.x < 66) ws[threadIdx.x] = 0.0f;
}

__global__ __launch_bounds__(128)
void router_main_kernel(const float* __restrict__ x,
                        const float* __restrict__ Wg,
                        float* __restrict__ out,
                        float* __restrict__ ws) {
    __shared__ float lgt[TOK_PER_BLOCK][N_EXP + 1]; // +1 pad: avoid LDS bank conflicts

    const int tid  = threadIdx.x;
    const int wave = tid >> 5;
    const int lane = tid & 31;
    const int lLo  = lane & 15;   // M (A) / N (B) index within 16
    const int lHi  = lane >> 4;   // selects K pair {0,1} vs {2,3}

    const int tokBase = blockIdx.x * TOK_PER_BLOCK + wave * 16;

    // A fragment source: x[m][k0 + 2*lHi + {0,1}]  (16x4 fp32 A layout)
    const float* ax = x + (size_t)(tokBase + lLo) * D_MODEL + (lHi << 1);
    // B fragment source: B[k][n] = Wg[n][k]; lane holds Wg[n][k0 + 2*lHi + {0,1}]
    const float* bw[4];
#pragma unroll
    for (int t = 0; t < 4; ++t)
        bw[t] = Wg + (size_t)(t * 16 + lLo) * D_MODEL + (lHi << 1);

    v8f acc[4];
#pragma unroll
    for (int t = 0; t < 4; ++t)
        acc[t] = (v8f){0.f, 0.f, 0.f, 0.f, 0.f, 0.f, 0.f, 0.f};

    // K loop: 2048 in steps of 4, unrolled 4x -> 16 v_wmma + 20 b64 loads / iter
#pragma unroll 4
    for (int k = 0; k < D_MODEL; k += 4) {
        v2f a = *(const v2f*)(ax + k);
#pragma unroll
        for (int t = 0; t < 4; ++t) {
            v2f b = *(const v2f*)(bw[t] + k);
            acc[t] = __builtin_amdgcn_wmma_f32_16x16x4_f32(
                /*neg_a=*/false, a, /*neg_b=*/false, b,
                /*c_mod=*/(short)0, acc[t], /*reuse_a=*/false, /*reuse_b=*/false);
        }
    }

    // C/D layout: VGPR r, lanes 0-15 -> M=r, lanes 16-31 -> M=r+8; N = lane%16
#pragma unroll
    for (int t = 0; t < 4; ++t) {
#pragma unroll
        for (int r = 0; r < 8; ++r) {
            lgt[wave * 16 + r + 8 * lHi][t * 16 + lLo] = acc[t][r];
        }
    }

    __syncthreads();

    // One thread per token: softmax over 64 experts, top-2, entropy, outputs
    if (tid < TOK_PER_BLOCK) {
        const int tok = blockIdx.x * TOK_PER_BLOCK + tid;
        const float* row = lgt[tid];

        float mx = row[0];
#pragma unroll
        for (int e = 1; e < N_EXP; ++e) mx = fmaxf(mx, row[e]);

        float s = 0.0f;
#pragma unroll
        for (int e = 0; e < N_EXP; ++e) s += __expf(row[e] - mx);
        const float inv = 1.0f / s;

        float ent = 0.0f;
        float l0 = -INFINITY, l1 = -INFINITY;
        int   i0 = 0, i1 = 0;
#pragma unroll
        for (int e = 0; e < N_EXP; ++e) {
            const float v = row[e];
            const float p = __expf(v - mx) * inv;
            out[OFF_P + (size_t)tok * N_EXP + e] = p;
            ent -= p * __logf(p + 1e-10f);
            if (v > l0)      { l1 = l0; i1 = i0; l0 = v; i0 = e; }
            else if (v > l1) { l1 = v; i1 = e; }
        }

        // softmax over the two top logits (l0 >= l1)
        const float w0 = 1.0f / (1.0f + __expf(l1 - l0));
        const float w1 = 1.0f - w0;

        out[OFF_W + tok * 2 + 0] = w0;
        out[OFF_W + tok * 2 + 1] = w1;
        out[OFF_I + tok * 2 + 0] = (float)i0;
        out[OFF_I + tok * 2 + 1] = (float)i1;

        atomicAdd(&ws[0], ent);
        atomicAdd(&ws[1], w0);
        atomicAdd(&ws[2 + i0], 1.0f);
        atomicAdd(&ws[2 + i1], 1.0f);
    }
}

__global__ void router_final_kernel(const float* __restrict__ ws,
                                    float* __restrict__ out) {
    const int t = threadIdx.x;
    if (t == 0) {
        out[OFF_ENT]  = ws[0] * (1.0f / (float)N_TOK);
        out[OFF_CONF] = ws[1] * (1.0f / (float)N_TOK);
    }
    if (t < N_EXP) {
        // counts sum is exactly 2*N_TOK every call
        out[OFF_UTIL + t] = ws[2 + t] * (1.0f / (2.0f * (float)N_TOK + 1e-10f));
    }
}

extern "C" void kernel_launch(void* const* d_in, const int* in_sizes, int n_in,
                              void* d_out, int out_size, void* d_ws, size_t ws_size,
                              hipStream_t stream) {
    const float* x  = (const float*)d_in[0];   // [4,4096,2048]
    const float* Wg = (const float*)d_in[1];   // [64,2048]
    float* out = (float*)d_out;
    float* ws  = (float*)d_ws;

    router_init_kernel<<<1, 128, 0, stream>>>(ws);
    router_main_kernel<<<N_TOK / TOK_PER_BLOCK, 128, 0, stream>>>(x, Wg, out, ws);
    router_final_kernel<<<1, 64, 0, stream>>>(ws, out);
}